// KVCacheCausalSelfAttention_18726057411073
// MI455X (gfx1250) — compile-verified
//
#include <hip/hip_runtime.h>

typedef __attribute__((ext_vector_type(2))) float v2f;
typedef __attribute__((ext_vector_type(8))) float v8f;

#define LOG2E 1.4426950408889634f

// ---------------------------------------------------------------------------
// GEMM: C[32 x N] = A[32 x K] * W[K x N] + bias[N]   (all fp32, row-major)
// One 16x16 tile of C per 256-thread block. 8 waves split the K range
// (in-block split-K); partial tiles reduced deterministically through LDS.
// K is a compile-time constant so the inner loop unrolls (8 wmmas per group,
// loads batched ahead of waits) and loop control stays wave-uniform.
// Uses V_WMMA_F32_16X16X4_F32 (full-precision CDNA5 WMMA).
// A fragment (16x4 f32): lane l<16 holds A[m0+l][k..k+1], l>=16 holds k+2..k+3.
// B fragment (4x16 f32): VGPR j, lane-half hh -> row K = 2*hh + j, col n0+(l&15).
// ---------------------------------------------------------------------------
template <int K>
__global__ __launch_bounds__(256)
void gemm_wmma_f32(const float* __restrict__ A, const float* __restrict__ W,
                   const float* __restrict__ bias, float* __restrict__ C,
                   int N) {
  constexpr int KS = K / 8;           // per-wave K slice (1024 -> 128)
  const int tid  = threadIdx.x;
  const int wave = tid >> 5;
  const int lane = tid & 31;
  const int half = lane >> 4;
  const int l15  = lane & 15;
  const int n0 = blockIdx.x * 16;
  const int m0 = blockIdx.y * 16;

  __shared__ float part[8 * 256];

  // wave's K offset folded into base pointers; loop bounds are compile-time
  const float* arow = A + (size_t)(m0 + l15) * K + wave * KS + 2 * half;
  const float* wb   = W + (size_t)(wave * KS + 2 * half) * N + n0 + l15;

  v8f c = {};
  #pragma unroll 8
  for (int k = 0; k < KS; k += 4) {
    v2f a = *(const v2f*)(arow + k);          // A[row][kb], A[row][kb+1]
    v2f b;
    b.x = wb[(size_t)k * N];                  // B[K=kb  ][n0+l15]
    b.y = wb[(size_t)k * N + N];              // B[K=kb+1][n0+l15]
    c = __builtin_amdgcn_wmma_f32_16x16x4_f32(false, a, false, b, (short)0, c,
                                              false, false);
  }

  // stage per-wave partial tile (row-major 16x16)
  #pragma unroll
  for (int i = 0; i < 8; ++i)
    part[wave * 256 + (half * 8 + i) * 16 + l15] = c[i];
  __syncthreads();

  // deterministic fixed-order reduction: one output element per thread
  const int row = tid >> 4, col = tid & 15;
  float v = 0.f;
  #pragma unroll
  for (int w = 0; w < 8; ++w) v += part[w * 256 + tid];
  C[(size_t)(m0 + row) * N + (n0 + col)] = v + bias[n0 + col];
}

// ---------------------------------------------------------------------------
// Flash-decode attention, one (b,h) per block, 8 waves of 32.
// qkv: [32][3072] = q | k_new | v_new.  kc/vc: [32][16][8192][64].
// New token's k/v row is read from qkv at position idx (caches not mutated).
// Online softmax per wave over 32-key chunks; deterministic cross-wave merge.
// ---------------------------------------------------------------------------
__global__ __launch_bounds__(256)
void attn_decode(const float* __restrict__ qkv, const float* __restrict__ kc,
                 const float* __restrict__ vc, float* __restrict__ yatt,
                 const int* __restrict__ idxp) {
  const int bh = blockIdx.x;          // b*16 + h
  const int b  = bh >> 4;
  const int h  = bh & 15;
  const int tid  = threadIdx.x;
  const int wave = tid >> 5;
  const int lane = tid & 31;
  const int idx = *idxp;
  const int Tg  = idx + 1;

  __shared__ float red_acc[8][64];
  __shared__ float red_m[8];
  __shared__ float red_l[8];

  const float* qp    = qkv + (size_t)b * 3072 + h * 64;
  const float* knew  = qp + 1024;
  const float* vnew  = qp + 2048;
  const float* kbase = kc + (size_t)bh * 8192 * 64;
  const float* vbase = vc + (size_t)bh * 8192 * 64;

  // q in registers, pre-scaled by 1/sqrt(hd) * log2(e) so exp2 == exp
  const float qs = 0.125f * LOG2E;
  float4 q4[16];
  #pragma unroll
  for (int i = 0; i < 16; ++i) {
    float4 t = ((const float4*)qp)[i];
    q4[i].x = t.x * qs; q4[i].y = t.y * qs;
    q4[i].z = t.z * qs; q4[i].w = t.w * qs;
  }

  float m = -3.0e38f;                 // running max (exp2 domain)
  float lsum = 0.f;                   // running denominator
  float accx = 0.f, accy = 0.f;       // this lane owns dims 2*lane, 2*lane+1

  const int nchunks = (Tg + 31) >> 5; // 129 for Tg=4097
  for (int c = wave; c < nchunks; c += 8) {
    const int j = c * 32 + lane;      // key index for this lane (<= 4127 < 8192)

    // prefetch next chunk's rows for this wave (stays inside the 8192-row
    // cache allocation: j+256 <= 4383). Streams HBM ahead of use.
    __builtin_prefetch(kbase + (size_t)(j + 256) * 64, 0, 0);
    __builtin_prefetch(vbase + (size_t)(j + 256) * 64, 0, 0);

    const float* kr = (j == idx) ? knew : (kbase + (size_t)j * 64);
    float s = 0.f;
    #pragma unroll
    for (int i = 0; i < 16; ++i) {
      float4 k4 = ((const float4*)kr)[i];
      s += q4[i].x * k4.x + q4[i].y * k4.y + q4[i].z * k4.z + q4[i].w * k4.w;
    }
    if (j >= Tg) s = -3.0e38f;        // mask lanes past the causal horizon

    // wave-wide max of this chunk
    float cm = s;
    #pragma unroll
    for (int o = 16; o >= 1; o >>= 1) cm = fmaxf(cm, __shfl_xor(cm, o, 32));
    const float mn = fmaxf(m, cm);    // cm always finite (every chunk has lane 0 valid)
    const float r = exp2f(m - mn);    // first chunk: exp2(-huge) = 0
    lsum *= r; accx *= r; accy *= r;
    const float p = exp2f(s - mn);    // masked lanes -> 0
    float ps = p;
    #pragma unroll
    for (int o = 16; o >= 1; o >>= 1) ps += __shfl_xor(ps, o, 32);
    lsum += ps;
    m = mn;

    // cooperative V accumulation: broadcast p, coalesced 256B v-row loads
    int limit = Tg - c * 32; if (limit > 32) limit = 32;
    for (int kk = 0; kk < limit; ++kk) {
      const float pj = __shfl(p, kk, 32);
      const int jj = c * 32 + kk;
      const float* vr = (jj == idx) ? vnew : (vbase + (size_t)jj * 64);
      const float2 v2 = ((const float2*)vr)[lane];
      accx += pj * v2.x; accy += pj * v2.y;
    }
  }

  if (lane == 0) { red_m[wave] = m; red_l[wave] = lsum; }
  red_acc[wave][2 * lane]     = accx;
  red_acc[wave][2 * lane + 1] = accy;
  __syncthreads();

  // deterministic cross-wave merge: one output dim per thread (tid < 64)
  if (tid < 64) {
    float M = red_m[0];
    #pragma unroll
    for (int w = 1; w < 8; ++w) M = fmaxf(M, red_m[w]);
    float L = 0.f, val = 0.f;
    #pragma unroll
    for (int w = 0; w < 8; ++w) {
      const float wg = exp2f(red_m[w] - M);
      L   += red_l[w] * wg;
      val += red_acc[w][tid] * wg;
    }
    yatt[(size_t)bh * 64 + tid] = val / L;   // y[b][h*64+d] layout == (B, C)
  }
}

extern "C" void kernel_launch(void* const* d_in, const int* in_sizes, int n_in,
                              void* d_out, int out_size, void* d_ws, size_t ws_size,
                              hipStream_t stream) {
  (void)in_sizes; (void)n_in; (void)out_size; (void)ws_size;
  const float* x      = (const float*)d_in[0];  // (32,1,1024)
  const float* kcache = (const float*)d_in[1];  // (32,16,8192,64)
  const float* vcache = (const float*)d_in[2];
  const float* W_attn = (const float*)d_in[3];  // (1024,3072)
  const float* b_attn = (const float*)d_in[4];
  const float* W_proj = (const float*)d_in[5];  // (1024,1024)
  const float* b_proj = (const float*)d_in[6];
  const int*   idxp   = (const int*)d_in[7];

  float* qkv  = (float*)d_ws;                   // 32*3072 floats
  float* yatt = (float*)d_ws + 32 * 3072;       // 32*1024 floats
  float* out  = (float*)d_out;                  // 32*1024 floats

  // 1) qkv = x @ W_attn + b_attn        (WMMA fp32)
  gemm_wmma_f32<1024><<<dim3(3072 / 16, 2), 256, 0, stream>>>(x, W_attn, b_attn,
                                                              qkv, 3072);
  // 2) flash-decode attention over the KV cache + new token
  attn_decode<<<dim3(512), 256, 0, stream>>>(qkv, kcache, vcache, yatt, idxp);
  // 3) y = yatt @ W_proj + b_proj       (WMMA fp32)
  gemm_wmma_f32<1024><<<dim3(1024 / 16, 2), 256, 0, stream>>>(yatt, W_proj,
                                                              b_proj, out, 1024);
}